// Net_26680336843646
// MI455X (gfx1250) — compile-verified
//
#include <hip/hip_runtime.h>

#define N_NODES   100000
#define N_EDGES   800000
#define N_GRAPHS  4000
#define EMB       128
#define HID       256
#define FPD       2048

typedef float v2f __attribute__((ext_vector_type(2)));
typedef float v8f __attribute__((ext_vector_type(8)));

#define KC   64          // K chunk staged in LDS
#define GB   256         // gemm block threads (8 waves)
#define LDSP 68          // padded LDS row stride: 16B-aligned rows, conflict-free cols

struct EmbGather {
  const float* t0; const float* t1; const float* t2;
  const int* i0; const int* i1; const int* i2;
};

__device__ __forceinline__ void atomAddF(float* p, float v) {
  unsafeAtomicAdd(p, v);   // HW global_atomic_add_f32
}

// ---------------------------------------------------------------------------
// f32 WMMA GEMM: C[M,N] = op(A[M,K]) * B[K,N] (+ bias) (+ C if accum) (relu?)
// grid.x = M/16 (M exact multiple of 16), grid.y = N/128.
// LDB (= N = ldb = ldc) is a template constant so all B-fragment loads are
// immediate-offset global loads off one base register; a whole 64-deep K
// chunk of fragments is register-staged, then 16 WMMAs issue back-to-back.
// GATHER=true: A row m is the concat of three embedding-table rows.
// ---------------------------------------------------------------------------
template<bool GATHER, int LDB>
__global__ __launch_bounds__(GB)
void gemm_wmma(const float* __restrict__ A, int lda,
               const float* __restrict__ B,
               const float* __restrict__ bias,
               float* __restrict__ C,
               int K, int accum, int relu, EmbGather g)
{
  __shared__ float As[16 * LDSP];
  const int tid  = threadIdx.x;
  const int lane = tid & 31;
  const int wave = tid >> 5;
  const int row0 = blockIdx.x * 16;
  const int col0 = blockIdx.y * 128 + wave * 16;
  const int ml   = lane & 15;          // M (A) / N (B,C) lane index
  const int kk   = (lane >> 4) * 2;    // K sub-offset for upper half-wave

  // staging coords: each thread moves one float4 per chunk
  const int sm = tid >> 4;             // row 0..15
  const int sk = (tid & 15) * 4;       // k offset 0..60

  v8f acc = {0.f, 0.f, 0.f, 0.f, 0.f, 0.f, 0.f, 0.f};

  // fold the lane-dependent kk*LDB into the base; per-chunk advance is += KC*LDB
  const float* Bp = B + col0 + ml + (long)kk * LDB;

  for (int kc = 0; kc < K; kc += KC) {
    __syncthreads();
    float4 av;
    if (GATHER) {
      int gk = kc + sk;                // float4 never straddles a 128-wide field
      int node = row0 + sm;
      const float* t; int fi;
      if      (gk < EMB)     { t = g.t0; fi = g.i0[node]; }
      else if (gk < 2 * EMB) { t = g.t1; fi = g.i1[node]; gk -= EMB; }
      else                   { t = g.t2; fi = g.i2[node]; gk -= 2 * EMB; }
      av = *(const float4*)(t + (long)fi * EMB + gk);
    } else {
      av = *(const float4*)(A + (long)(row0 + sm) * lda + kc + sk);
    }
    *(float4*)&As[sm * LDSP + sk] = av;
    __syncthreads();

    // register-stage all fragments for this chunk (immediate-offset loads)
    v2f bf[16];
#pragma unroll
    for (int j = 0; j < 16; ++j) {
      bf[j].x = Bp[4 * j * LDB];
      bf[j].y = Bp[(4 * j + 1) * LDB];
    }
    v2f af[16];
#pragma unroll
    for (int j = 0; j < 16; ++j) {
      af[j].x = As[ml * LDSP + 4 * j + kk];
      af[j].y = As[ml * LDSP + 4 * j + kk + 1];
    }
#pragma unroll
    for (int j = 0; j < 16; ++j)
      acc = __builtin_amdgcn_wmma_f32_16x16x4_f32(
          false, af[j], false, bf[j], (short)0, acc, false, false);

    Bp += (long)KC * LDB;
  }

  // C/D 16x16 frag: VGPR r -> row r (lanes 0-15) / row r+8 (lanes 16-31)
  const int   coln = col0 + ml;
  const float bv   = bias ? bias[coln] : 0.0f;
  const int   rb   = (lane >> 4) * 8;
#pragma unroll
  for (int r = 0; r < 8; ++r) {
    long idx = (long)(row0 + rb + r) * LDB + coln;
    float v = acc[r] + bv;
    if (accum) v += C[idx];
    if (relu)  v = fmaxf(v, 0.0f);
    C[idx] = v;
  }
}

// ---------------------------------------------------------------------------
// Elementwise / scatter kernels
// ---------------------------------------------------------------------------
__global__ void k_fill(float* __restrict__ p, float v, long n) {
  long i = blockIdx.x * (long)blockDim.x + threadIdx.x;
  if (i < n) p[i] = v;
}

__global__ void k_deg_count(const int* __restrict__ dst, float* __restrict__ deg, int E) {
  int e = blockIdx.x * blockDim.x + threadIdx.x;
  if (e < E) atomAddF(&deg[dst[e]], 1.0f);
}

__global__ void k_dinv(const float* __restrict__ deg, float* __restrict__ dinv, int n) {
  int i = blockIdx.x * blockDim.x + threadIdx.x;
  if (i < n) dinv[i] = rsqrtf(deg[i]);
}

// Y[n,:] = X[n,:] * dinv[n]^2  (self-loop term), float4 over n*HID/4 elems
__global__ void k_self_init(const float* __restrict__ X, const float* __restrict__ dinv,
                            float* __restrict__ Y, long n4) {
  long i = blockIdx.x * (long)blockDim.x + threadIdx.x;
  if (i >= n4) return;
  int node = (int)(i >> 6);            // HID/4 == 64 float4 per row
  float d = dinv[node]; d = d * d;
  float4 x = ((const float4*)X)[i];
  float4 y = make_float4(x.x * d, x.y * d, x.z * d, x.w * d);
  ((float4*)Y)[i] = y;
}

// one wave per edge: Y[dst,:] += X[src,:] * dinv[src]*dinv[dst]
__global__ void k_edge_scatter(const int* __restrict__ src, const int* __restrict__ dst,
                               const float* __restrict__ dinv,
                               const float* __restrict__ X, float* __restrict__ Y, int E) {
  int wid  = (blockIdx.x * blockDim.x + threadIdx.x) >> 5;
  int lane = threadIdx.x & 31;
  if (wid >= E) return;
  int s = src[wid], d = dst[wid];
  float norm = dinv[s] * dinv[d];
  const float4* xs = (const float4*)(X + (long)s * HID);
  float*        yd = Y + (long)d * HID;
#pragma unroll
  for (int j = 0; j < 2; ++j) {
    int c4 = lane + j * 32;            // 64 float4 per row
    float4 v = xs[c4];
    atomAddF(&yd[c4 * 4 + 0], v.x * norm);
    atomAddF(&yd[c4 * 4 + 1], v.y * norm);
    atomAddF(&yd[c4 * 4 + 2], v.z * norm);
    atomAddF(&yd[c4 * 4 + 3], v.w * norm);
  }
}

// H = relu(Y + bias), float4 vectorized
__global__ void k_bias_relu(const float* __restrict__ Y, const float* __restrict__ b,
                            float* __restrict__ H, long n4) {
  long i = blockIdx.x * (long)blockDim.x + threadIdx.x;
  if (i >= n4) return;
  float4 y  = ((const float4*)Y)[i];
  float4 bb = ((const float4*)b)[i & 63];
  float4 h = make_float4(fmaxf(y.x + bb.x, 0.f), fmaxf(y.y + bb.y, 0.f),
                         fmaxf(y.z + bb.z, 0.f), fmaxf(y.w + bb.w, 0.f));
  ((float4*)H)[i] = h;
}

// one wave per node: pooled[batch[n],:] += H[n,:]
__global__ void k_pool(const float* __restrict__ H, const int* __restrict__ batch,
                       float* __restrict__ pooled, int n) {
  int wid  = (blockIdx.x * blockDim.x + threadIdx.x) >> 5;
  int lane = threadIdx.x & 31;
  if (wid >= n) return;
  int gph = batch[wid];
  const float4* hs = (const float4*)(H + (long)wid * HID);
  float*        pd = pooled + (long)gph * HID;
#pragma unroll
  for (int j = 0; j < 2; ++j) {
    int c4 = lane + j * 32;
    float4 v = hs[c4];
    atomAddF(&pd[c4 * 4 + 0], v.x);
    atomAddF(&pd[c4 * 4 + 1], v.y);
    atomAddF(&pd[c4 * 4 + 2], v.z);
    atomAddF(&pd[c4 * 4 + 3], v.w);
  }
}

// final GEMV: out[g] = z1[g,:] . w + b
__global__ void k_lin2(const float* __restrict__ z, const float* __restrict__ w,
                       const float* __restrict__ b, float* __restrict__ out, int n) {
  int gid = blockIdx.x * blockDim.x + threadIdx.x;
  if (gid >= n) return;
  const float4* zp = (const float4*)(z + (long)gid * 128);
  const float4* wp = (const float4*)w;
  float s = 0.f;
#pragma unroll 8
  for (int i = 0; i < 32; ++i) {
    float4 a = zp[i], c = wp[i];
    s += a.x * c.x + a.y * c.y + a.z * c.z + a.w * c.w;
  }
  out[gid] = s + b[0];
}

// ---------------------------------------------------------------------------
extern "C" void kernel_launch(void* const* d_in, const int* in_sizes, int n_in,
                              void* d_out, int out_size, void* d_ws, size_t ws_size,
                              hipStream_t stream) {
  const int*   atom_idx = (const int*)d_in[0];
  const int*   degr_idx = (const int*)d_in[1];
  const int*   chg_idx  = (const int*)d_in[2];
  const int*   edges    = (const int*)d_in[3];
  const int*   batch    = (const int*)d_in[4];
  const float* fng      = (const float*)d_in[5];
  const float* emb_a    = (const float*)d_in[6];
  const float* emb_d    = (const float*)d_in[7];
  const float* emb_c    = (const float*)d_in[8];
  const float* pw1 = (const float*)d_in[9];  const float* pb1 = (const float*)d_in[10];
  const float* pw2 = (const float*)d_in[11]; const float* pb2 = (const float*)d_in[12];
  const float* fpw = (const float*)d_in[13]; const float* fpb = (const float*)d_in[14];
  const float* cw[4] = {(const float*)d_in[15], (const float*)d_in[17],
                        (const float*)d_in[19], (const float*)d_in[21]};
  const float* cb[4] = {(const float*)d_in[16], (const float*)d_in[18],
                        (const float*)d_in[20], (const float*)d_in[22]};
  const float* l1w = (const float*)d_in[23]; const float* l1b = (const float*)d_in[24];
  const float* l2w = (const float*)d_in[25]; const float* l2b = (const float*)d_in[26];
  const int* srcE = edges;
  const int* dstE = edges + N_EDGES;

  // workspace layout (floats); bufP ([N,128] proj output) aliases buf2:
  // it is dead before buf2 is first written (stream-ordered).
  float* ws     = (float*)d_ws;
  float* buf1   = ws;                                   // N_NODES*HID
  float* buf2   = buf1 + (size_t)N_NODES * HID;         // N_NODES*HID
  float* bufP   = buf2;                                 // [N_NODES,128] alias
  float* deg    = buf2 + (size_t)N_NODES * HID;
  float* dinv   = deg + N_NODES;
  float* pooled = dinv + N_NODES;                       // N_GRAPHS*HID
  float* fpe    = pooled + (size_t)N_GRAPHS * HID;      // N_GRAPHS*HID
  float* z1     = fpe + (size_t)N_GRAPHS * HID;         // N_GRAPHS*128

  EmbGather g  {emb_a, emb_d, emb_c, atom_idx, degr_idx, chg_idx};
  EmbGather g0 {nullptr, nullptr, nullptr, nullptr, nullptr, nullptr};

  // degrees with self-loops: deg = 1 + indeg;  dinv = rsqrt(deg)
  k_fill<<<(N_NODES + 255) / 256, 256, 0, stream>>>(deg, 1.0f, (long)N_NODES);
  k_deg_count<<<(N_EDGES + 255) / 256, 256, 0, stream>>>(dstE, deg, N_EDGES);
  k_dinv<<<(N_NODES + 255) / 256, 256, 0, stream>>>(deg, dinv, N_NODES);

  // FFM projector: h1 = relu(gathered_emb @ W1 + b1); h = h1 @ W2 + b2
  gemm_wmma<true , 128><<<dim3(N_NODES / 16, 1), GB, 0, stream>>>(
      nullptr, 0, pw1, pb1, buf1, 3 * EMB, 0, 1, g);
  gemm_wmma<false, 128><<<dim3(N_NODES / 16, 1), GB, 0, stream>>>(
      buf1, EMB, pw2, pb2, bufP, EMB, 0, 0, g0);

  // 4 GCN layers, ping-pong buffers
  const float* h = bufP; int lda = EMB; int K = EMB;
  const long n4 = (long)N_NODES * (HID / 4);
  for (int l = 0; l < 4; ++l) {
    float* Xb = (l & 1) ? buf2 : buf1;   // xw = h @ W
    float* Yb = (l & 1) ? buf1 : buf2;   // aggregation buffer
    gemm_wmma<false, 256><<<dim3(N_NODES / 16, 2), GB, 0, stream>>>(
        h, lda, cw[l], nullptr, Xb, K, 0, 0, g0);
    k_self_init<<<(n4 + 255) / 256, 256, 0, stream>>>(Xb, dinv, Yb, n4);
    k_edge_scatter<<<(N_EDGES + 7) / 8, 256, 0, stream>>>(srcE, dstE, dinv, Xb, Yb, N_EDGES);
    k_bias_relu<<<(n4 + 255) / 256, 256, 0, stream>>>(Yb, cb[l], Xb, n4);
    h = Xb; lda = HID; K = HID;
  }

  // scatter-sum pooling
  k_fill<<<((long)N_GRAPHS * HID + 255) / 256, 256, 0, stream>>>(
      pooled, 0.0f, (long)N_GRAPHS * HID);
  k_pool<<<(N_NODES + 7) / 8, 256, 0, stream>>>(h, batch, pooled, N_NODES);

  // fingerprint linear
  gemm_wmma<false, 256><<<dim3(N_GRAPHS / 16, 2), GB, 0, stream>>>(
      fng, FPD, fpw, fpb, fpe, FPD, 0, 0, g0);

  // head: z1 = relu([fpe, pooled] @ l1w + l1b) as two accumulated GEMMs
  gemm_wmma<false, 128><<<dim3(N_GRAPHS / 16, 1), GB, 0, stream>>>(
      fpe, HID, l1w, nullptr, z1, HID, 0, 0, g0);
  gemm_wmma<false, 128><<<dim3(N_GRAPHS / 16, 1), GB, 0, stream>>>(
      pooled, HID, l1w + 256 * 128, l1b, z1, HID, 1, 1, g0);

  // final GEMV
  k_lin2<<<(N_GRAPHS + 255) / 256, 256, 0, stream>>>(z1, l2w, l2b, (float*)d_out, N_GRAPHS);
}